// LSTM_19516331393564
// MI455X (gfx1250) — compile-verified
//
#include <hip/hip_runtime.h>
#include <hip/hip_bf16.h>

typedef __attribute__((ext_vector_type(2))) float v2f;
typedef __attribute__((ext_vector_type(8))) float v8f;
typedef __attribute__((ext_vector_type(4))) unsigned int u32x4;
typedef __attribute__((ext_vector_type(8))) int i32x8;
typedef __attribute__((ext_vector_type(4))) int i32x4;

#define BB   8
#define TT   512
#define FF   1024
#define GG   (4 * FF)     // 4096 gate width
#define VV   32000
#define KC   256          // K-chunk staged per TDM transfer (16 rows x 256 f32 = 16KB)

// ---------------------------------------------------------------------------
// TDM descriptor issue: 2D tile load (16 rows x tile_k f32) Global -> LDS.
// D# layout per CDNA5 ISA ch.8: group0 = control/lds/global addr,
// group1 = dims/strides, groups 2-3 = row indices (gather) or zero.
// This toolchain's builtin takes the 6-arg form:
//   (u32x4 g0, i32x8 g1, i32x4 g2, i32x4 g3, i32x8 extra, i32 cpol)
// ---------------------------------------------------------------------------
__device__ __forceinline__ void tdm_issue(unsigned lds_off,
                                          const float* gaddr,
                                          unsigned tile_k,
                                          unsigned rows,
                                          unsigned tensor_k,
                                          unsigned row_stride_elems,
                                          bool gather,
                                          i32x4 idx_lo, i32x4 idx_hi)
{
    const unsigned long long ga = (unsigned long long)(uintptr_t)gaddr;
    u32x4 g0;
    g0.x = gather ? (1u | (1u << 31)) : 1u;     // count=1; gather_mode; 16-bit idx
    g0.y = lds_off;                              // LDS byte address
    g0.z = (unsigned)(ga & 0xFFFFFFFFu);         // global_addr[31:0]
    g0.w = (unsigned)((ga >> 32) & 0x01FFFFFFu) | (2u << 30);  // [56:32] | type=2

    i32x8 g1;
    g1[0] = (int)(2u << 16);                                     // data_size=4B
    g1[1] = (int)((tensor_k & 0xFFFFu) << 16);                   // tensor_dim0 lo
    g1[2] = (int)(((tensor_k >> 16) & 0xFFFFu) |
                  ((rows & 0xFFFFu) << 16));                     // td0 hi | td1 lo
    g1[3] = (int)(((rows >> 16) & 0xFFFFu) |
                  ((tile_k & 0xFFFFu) << 16));                   // td1 hi | tile_dim0
    g1[4] = (int)(rows & 0xFFFFu);                               // tile_dim1 (#rows / #indices)
    g1[5] = (int)row_stride_elems;                               // tensor_dim0_stride lo
    g1[6] = 0;
    g1[7] = 0;

    const i32x8 extra = {0, 0, 0, 0, 0, 0, 0, 0};
    __builtin_amdgcn_tensor_load_to_lds(g0, g1, idx_lo, idx_hi, extra, 0);
}

// ---------------------------------------------------------------------------
// fp32 WMMA GEMM:  C[m,n] = sum_k A[m,k]*B[k,n] + bias[n]
//   WG = 128 threads (4 waves); tile 16(M) x 64(N); each wave owns a 16x16
//   N-subtile and runs K/4 V_WMMA_F32_16X16X4_F32.
//   A staged in LDS by the Tensor Data Mover in KC-chunks, double buffered;
//   wave 0 issues the DMA, s_wait_tensorcnt + barrier publishes it.
//   If tokens != nullptr, the TDM runs in gather mode with the 16 token ids
//   of this M-strip as 16-bit row indices (fused embedding gather).
// ---------------------------------------------------------------------------
__global__ __launch_bounds__(128) void gemm_f32_wmma(
    const float* __restrict__ A,
    const int*   __restrict__ tokens,   // nullable
    const float* __restrict__ B,
    const float* __restrict__ bias,
    float*       __restrict__ C,
    int M, int N, int K)
{
    __shared__ float As[2][16 * KC];    // 32 KB double buffer

    const int tid  = threadIdx.x;
    const int m0   = blockIdx.x * 16;
    const int n0   = blockIdx.y * 64;
    const int nch  = K / KC;

    // Pack 16 row indices (wave-uniform scalar loads) for gather mode.
    i32x4 ilo = {0, 0, 0, 0}, ihi = {0, 0, 0, 0};
    const bool gather = (tokens != nullptr);
    if (gather) {
        #pragma unroll
        for (int r = 0; r < 4; ++r) {
            ilo[r] = (tokens[m0 + 2 * r] & 0xFFFF) |
                     (tokens[m0 + 2 * r + 1] << 16);
            ihi[r] = (tokens[m0 + 8 + 2 * r] & 0xFFFF) |
                     (tokens[m0 + 8 + 2 * r + 1] << 16);
        }
    }

    const unsigned lds_base = (unsigned)(uintptr_t)(&As[0][0]);
    const float* Abase = gather ? (A /* embed table */)
                                : (A + (long)m0 * (long)K);

    // Prologue: kick off chunk 0 into buffer 0 (wave 0 only; TDM is per-wave).
    if (tid < 32) {
        tdm_issue(lds_base, Abase /* +k offset 0 */, KC, 16, (unsigned)K,
                  (unsigned)K, gather, ilo, ihi);
    }

    const int lane = tid & 31;
    const int wave = tid >> 5;
    const int row  = lane & 15;          // M within tile (A side)
    const int klo  = (lane >> 4) << 1;   // 0 or 2
    const int nn   = n0 + wave * 16 + (lane & 15);
    const float* __restrict__ Bp = B + nn;

    v8f acc = {};
    for (int ch = 0; ch < nch; ++ch) {
        const int p = ch & 1;
        if (tid < 32) {
            if (ch + 1 < nch) {
                // buffer p^1 was released by the previous iteration's barrier
                tdm_issue(lds_base + (unsigned)((p ^ 1) * 16 * KC * 4),
                          Abase + (long)(ch + 1) * KC, KC, 16, (unsigned)K,
                          (unsigned)K, gather, ilo, ihi);
                __builtin_amdgcn_s_wait_tensorcnt(1);   // chunk ch landed
            } else {
                __builtin_amdgcn_s_wait_tensorcnt(0);
            }
        }
        __syncthreads();                                // publish buffer p

        const float* __restrict__ Ap = &As[p][0];
        const int kg0 = ch * KC;
        for (int kb = 0; kb < KC; kb += 4) {
            v2f a;
            a.x = Ap[row * KC + kb + klo];
            a.y = Ap[row * KC + kb + klo + 1];
            v2f bfrag;
            bfrag.x = Bp[(long)(kg0 + kb + klo)     * (long)N];
            bfrag.y = Bp[(long)(kg0 + kb + klo + 1) * (long)N];
            acc = __builtin_amdgcn_wmma_f32_16x16x4_f32(
                false, a, false, bfrag, (short)0, acc, false, false);
        }
        __syncthreads();                                // buffer p reusable
    }

    const float bs = bias[nn];
    const int mbase = (lane >> 4) * 8;   // lanes 16-31 hold M = 8..15
    #pragma unroll
    for (int r8 = 0; r8 < 8; ++r8) {
        C[(long)(m0 + mbase + r8) * (long)N + nn] = acc[r8] + bs;
    }
}

// ---------------------------------------------------------------------------
// One LSTM timestep, fully fused (recurrent matvec + gates + state update).
//   grid = 64 WGs x 128 threads; WG owns 16 f-columns across all 4 gates, so
//   the nonlinearity + c/h update happen in-kernel (no grid sync).
//   h ping-pongs across steps.
// ---------------------------------------------------------------------------
__global__ __launch_bounds__(128) void lstm_step(
    const float* __restrict__ xW,    // [B*T, 4F], m = b*T + t
    const float* __restrict__ Wh,    // [F, 4F]
    const float* __restrict__ h_in,  // [B, F]
    float*       __restrict__ h_out, // [B, F]
    float*       __restrict__ c,     // [B, F]
    float*       __restrict__ hs,    // [B, T, F]
    int t)
{
    __shared__ float hsh[BB * FF];   // 32 KB

    const int tid = threadIdx.x;
    for (int i = tid; i < BB * FF; i += 128) hsh[i] = h_in[i];
    __syncthreads();

    const int b  = tid >> 4;
    const int fi = tid & 15;
    const int f  = blockIdx.x * 16 + fi;

    const float* __restrict__ hrow = &hsh[b * FF];
    const float* __restrict__ wcol = Wh + f;

    float ai = 0.f, af = 0.f, ag = 0.f, ao = 0.f;
    for (int k = 0; k < FF; ++k) {
        const float hv = hrow[k];
        const float* __restrict__ wk = wcol + (long)k * GG;
        ai = fmaf(hv, wk[0],      ai);
        af = fmaf(hv, wk[FF],     af);
        ag = fmaf(hv, wk[2 * FF], ag);
        ao = fmaf(hv, wk[3 * FF], ao);
    }

    const long mrow = (long)(b * TT + t) * (long)GG;
    ai += xW[mrow + f];
    af += xW[mrow + FF + f];
    ag += xW[mrow + 2 * FF + f];
    ao += xW[mrow + 3 * FF + f];

    const float si = 1.f / (1.f + __expf(-ai));
    const float sf = 1.f / (1.f + __expf(-af));
    const float so = 1.f / (1.f + __expf(-ao));
    const float tg = tanhf(ag);

    const int cf = b * FF + f;
    const float cn = sf * c[cf] + si * tg;
    c[cf] = cn;
    const float hn = so * tanhf(cn);
    h_out[cf] = hn;
    hs[((long)b * TT + t) * (long)FF + f] = hn;
}

__global__ void init_state(float* __restrict__ h0,
                           float* __restrict__ h1,
                           float* __restrict__ c)
{
    const int i = blockIdx.x * blockDim.x + threadIdx.x;
    if (i < BB * FF) { h0[i] = 0.f; h1[i] = 0.f; c[i] = 0.f; }
}

// ---------------------------------------------------------------------------
extern "C" void kernel_launch(void* const* d_in, const int* in_sizes, int n_in,
                              void* d_out, int out_size, void* d_ws, size_t ws_size,
                              hipStream_t stream) {
    const int*   tokens = (const int*)  d_in[0];  // [B,T]
    const float* embed  = (const float*)d_in[1];  // [V+1, F]
    const float* Wx     = (const float*)d_in[2];  // [F, 4F]
    const float* Wh     = (const float*)d_in[3];  // [F, 4F]
    const float* bg     = (const float*)d_in[4];  // [4F]
    const float* Wd     = (const float*)d_in[5];  // [F, V]
    const float* bd     = (const float*)d_in[6];  // [V]
    float* out = (float*)d_out;                   // [B, T, V]

    // Workspace carving (~84 MB)
    char* ws = (char*)d_ws;
    float* xW = (float*)ws;                         ws += (size_t)BB * TT * GG * 4;  // 64 MB
    float* hs = (float*)ws;                         ws += (size_t)BB * TT * FF * 4;  // 16 MB
    float* h0 = (float*)ws;                         ws += (size_t)BB * FF * 4;
    float* h1 = (float*)ws;                         ws += (size_t)BB * FF * 4;
    float* cc = (float*)ws;                         ws += (size_t)BB * FF * 4;

    // 1) zero recurrent state
    init_state<<<(BB * FF + 255) / 256, 256, 0, stream>>>(h0, h1, cc);

    // 2) xW = embed[tokens] @ Wx + b   (TDM row-gather + fp32 WMMA)
    {
        dim3 grid((BB * TT) / 16, GG / 64);
        gemm_f32_wmma<<<grid, 128, 0, stream>>>(embed, tokens, Wx, bg, xW,
                                                BB * TT, GG, FF);
    }

    // 3) sequential scan, ping-ponging h
    for (int t = 0; t < TT; ++t) {
        const float* hin  = (t & 1) ? h1 : h0;
        float*       hout = (t & 1) ? h0 : h1;
        lstm_step<<<FF / 16, 128, 0, stream>>>(xW, Wh, hin, hout, cc, hs, t);
    }

    // 4) logits = hs @ Wd + bd   (TDM-staged A + fp32 WMMA, ~268 GFLOP)
    {
        dim3 grid((BB * TT) / 16, VV / 64);
        gemm_f32_wmma<<<grid, 128, 0, stream>>>(hs, nullptr, Wd, bd, out,
                                                BB * TT, VV, FF);
    }
}